// GCNLayer_89240830476477
// MI455X (gfx1250) — compile-verified
//
#include <hip/hip_runtime.h>

// CDNA5 wave32 WMMA fragment types
typedef __attribute__((ext_vector_type(2))) float v2f;
typedef __attribute__((ext_vector_type(8))) float v8f;

#define D 256           // D_IN == D_OUT == 256
#define PAIR_STRIDE 272 // float2 per K-pair row: 256 cols + 16 pad.
                        // = 544 dwords; 544 mod 64 == 32 -> half-lane group 1
                        // (+1 pair row) uses banks 32..63 while group 0 uses
                        // 0..31: a wave's b64 fragment load covers all 64
                        // banks exactly once (conflict-free).
                        // Total: 128 * 272 * 8 = 278528 B < 320KB WGP LDS.

extern __shared__ v2f ldsP[];  // dynamic LDS: [128 K-pairs][PAIR_STRIDE] v2f

// ---------------------------------------------------------------------------
// Kernel 1: y = x @ W   (fp32 GEMM via V_WMMA_F32_16X16X4_F32)
// W staged into LDS in *fragment order*: ldsP[p][c] = (W[2p][c], W[2p+1][c]).
// The v2f element type makes every fragment read a single 8B-aligned
// ds_load_b64 (16-bit byte offset -> almost no per-k address VALU).
// Persistent blocks: 256 thr = 8 waves; wave w: rowTileLocal = w>>2,
// colGroup = (w&3)*64. All control flow wave-uniform: EXEC all-1s for WMMA.
// ---------------------------------------------------------------------------
__global__ __launch_bounds__(256) void gcn_gemm_xw(
    const float* __restrict__ x, const float* __restrict__ W,
    float* __restrict__ y, int nRowTiles) {
  const int tid  = threadIdx.x;
  const int lane = tid & 31;
  const int wv   = tid >> 5;        // 0..7
  const int lr   = lane & 15;
  const int hi   = lane >> 4;       // which 16-lane half

  // ---- Stage W -> ldsP in K-pair (fragment) order ----
  // Pair (p, c) = (W[2p][c], W[2p+1][c]); lanes write consecutive c ->
  // ds_store_b64 with lane step 2 dwords -> all 64 banks, conflict-free.
#pragma unroll 4
  for (int g = tid; g < (D / 2) * D; g += 256) {
    const int p = g >> 8;       // K-pair index 0..127
    const int c = g & 255;      // column
    v2f val;
    val.x = W[(2 * p) * D + c];       // coalesced b32
    val.y = W[(2 * p + 1) * D + c];   // coalesced b32
    ldsP[p * PAIR_STRIDE + c] = val;  // ds_store_b64
  }
  __syncthreads();

  const int rowTileLocal = wv >> 2;       // 0 or 1
  const int colBase      = (wv & 3) * 64; // 0,64,128,192

  // Per-lane constant fragment base (in v2f units): pair row +hi, column.
  const int lbase = hi * PAIR_STRIDE + colBase + lr;

  for (int t = blockIdx.x * 2 + rowTileLocal; t < nRowTiles;
       t += gridDim.x * 2) {
    const int row0 = t * 16;

    v8f acc0 = {}, acc1 = {}, acc2 = {}, acc3 = {};

    // A-matrix 16x4 f32 layout: VGPR0 = K(2*hi), VGPR1 = K(2*hi+1)
    const float* xp = x + (row0 + lr) * D + 2 * hi;

    for (int k = 0; k < D; k += 4) {
      v2f a = *(const v2f*)(xp + k);        // global_load_b64, 8B aligned

      const int lb = lbase + (k >> 1) * PAIR_STRIDE;
      // One ds_load_b64 per fragment: (W[k+2hi][c], W[k+2hi+1][c])
      v2f b0 = ldsP[lb +  0];   // cols colBase + 0..15
      v2f b1 = ldsP[lb + 16];   // cols colBase + 16..31
      v2f b2 = ldsP[lb + 32];   // cols colBase + 32..47
      v2f b3 = ldsP[lb + 48];   // cols colBase + 48..63

      acc0 = __builtin_amdgcn_wmma_f32_16x16x4_f32(false, a, false, b0,
                                                   (short)0, acc0, false, false);
      acc1 = __builtin_amdgcn_wmma_f32_16x16x4_f32(false, a, false, b1,
                                                   (short)0, acc1, false, false);
      acc2 = __builtin_amdgcn_wmma_f32_16x16x4_f32(false, a, false, b2,
                                                   (short)0, acc2, false, false);
      acc3 = __builtin_amdgcn_wmma_f32_16x16x4_f32(false, a, false, b3,
                                                   (short)0, acc3, false, false);
    }

    // D-matrix layout: VGPR i holds row (i + 8*hi), col = lane & 15
    float* yp = y + row0 * D + colBase + lr;
#pragma unroll
    for (int i = 0; i < 8; ++i) {
      const int r = (i + 8 * hi) * D;
      yp[r +  0] = acc0[i];
      yp[r + 16] = acc1[i];
      yp[r + 32] = acc2[i];
      yp[r + 48] = acc3[i];
    }
  }
}

// ---------------------------------------------------------------------------
// Kernel 2: out[i][j] = b[j]   (bias broadcast; scatter then accumulates)
// ---------------------------------------------------------------------------
__global__ __launch_bounds__(256) void gcn_bias_init(
    const float* __restrict__ b, float* __restrict__ out) {
  const int i = blockIdx.x * D + threadIdx.x;
  out[i] = b[threadIdx.x];
}

// ---------------------------------------------------------------------------
// Kernel 3: per-edge scatter-add:  out[row[e]] += val[e] * y[col[e]]
// One wave per edge; each lane: two float4 gathers + eight
// global_atomic_add_f32. Both y (51MB) and out (51MB) are resident in the
// 192MB L2, so this is L2 atomic bandwidth, not HBM.
// ---------------------------------------------------------------------------
__global__ __launch_bounds__(256) void gcn_scatter(
    const int* __restrict__ erow, const int* __restrict__ ecol,
    const float* __restrict__ eval_, const float* __restrict__ y,
    float* __restrict__ out, int E) {
  const int lane = threadIdx.x & 31;
  const int e = blockIdx.x * 8 + (threadIdx.x >> 5);
  if (e >= E) return;

  const int   r = erow[e];
  const int   c = ecol[e];
  const float v = eval_[e];

  const float4* src = (const float4*)(y + (size_t)c * D);
  float* dst = out + (size_t)r * D;

#pragma unroll
  for (int it = 0; it < 2; ++it) {
    const int q = lane + 32 * it;  // float4 index 0..63
    float4 d = src[q];
    const int base = q * 4;
    atomicAdd(dst + base + 0, v * d.x);
    atomicAdd(dst + base + 1, v * d.y);
    atomicAdd(dst + base + 2, v * d.z);
    atomicAdd(dst + base + 3, v * d.w);
  }
}

// ---------------------------------------------------------------------------
// Launch: y = x@W  (WMMA, into d_ws);  out = bias;  out += scatter(val * y)
// Uses linearity: segsum(val*x[col]) @ W + b == segsum(val*(x@W)[col]) + b
// ---------------------------------------------------------------------------
extern "C" void kernel_launch(void* const* d_in, const int* in_sizes, int n_in,
                              void* d_out, int out_size, void* d_ws, size_t ws_size,
                              hipStream_t stream) {
  const float* x    = (const float*)d_in[0];
  const int*   erow = (const int*)d_in[1];
  const int*   ecol = (const int*)d_in[2];
  const float* ev   = (const float*)d_in[3];
  const float* W    = (const float*)d_in[4];
  const float* b    = (const float*)d_in[5];
  float*       out  = (float*)d_out;
  float*       y    = (float*)d_ws;          // N*D floats of scratch

  const int N = in_sizes[0] / D;             // 50000
  const int E = in_sizes[1];                 // 800000

  // 1) y = x @ W : persistent blocks, W staged in LDS in fragment order
  const size_t ldsBytes = (size_t)128 * PAIR_STRIDE * sizeof(v2f); // 278528
  gcn_gemm_xw<<<512, 256, ldsBytes, stream>>>(x, W, y, N / 16);

  // 2) out = broadcast(bias)
  gcn_bias_init<<<N, D, 0, stream>>>(b, out);

  // 3) out[row] += val * y[col]   (atomic scatter, 8 edges per block)
  gcn_scatter<<<(E + 7) / 8, 256, 0, stream>>>(erow, ecol, ev, y, out, E);
}